// GSPNLayer_34136400069261
// MI455X (gfx1250) — compile-verified
//
#include <hip/hip_runtime.h>

// GSPN layer for MI455X (gfx1250), wave32 + V_WMMA_F32_16X16X4_F32.
// B=8, C=256, Cr=64, S=64, P=4096.

#define Bsz 8
#define Cch 256
#define CR  64
#define Sdim 64
#define Pix 4096

typedef float v2f __attribute__((ext_vector_type(2)));
typedef float v8f __attribute__((ext_vector_type(8)));
typedef int   v4i __attribute__((ext_vector_type(4)));

#if defined(__gfx1250__) && __has_builtin(__builtin_amdgcn_global_load_async_to_lds_b128)
#define USE_ASYNC_LDS 1
#endif

__device__ __forceinline__ v8f wmma4(v2f a, v2f b, v8f c) {
  // D = A(16x4,f32) * B(4x16,f32) + C(16x16,f32)
  return __builtin_amdgcn_wmma_f32_16x16x4_f32(false, a, false, b, (short)0, c,
                                               false, false);
}

// ---------------------------------------------------------------- K0: fold merge weights
__global__ void k_prep_merge(const float* __restrict__ wm,
                             float* __restrict__ WA, float* __restrict__ WB) {
  int i = blockIdx.x * blockDim.x + threadIdx.x;   // 65536
  int o = i >> 8, c = i & 255;
  const float* r = wm + o * 1024;
  WA[i] = r[c] + r[c + 512];
  WB[i] = r[c + 256] + r[c + 768];
}

// ---------------------------------------------------------------- K1: reduced = w_red @ x
// reduced stored channel-major: red[b][cr][p]
__global__ void k_reduce(const float* __restrict__ x, const float* __restrict__ w_red,
                         const float* __restrict__ b_red, float* __restrict__ red) {
  const int lane = threadIdx.x & 31;
  const int mt   = threadIdx.x >> 5;       // 0..3  (cr tile)
  const int b    = blockIdx.x >> 6;        // 0..7
  const int nblk = blockIdx.x & 63;        // 64-pixel block
  const int row  = lane & 15;
  const int half = lane >> 4;
  const int koff = half * 2;
  const int m0 = mt * 16;
  const int n0 = nblk * 64;
  const float* X = x + (size_t)b * Cch * Pix;
  v8f acc[4] = {};
  for (int kk = 0; kk < Cch; kk += 4) {
    v2f a;
    a.x = w_red[(m0 + row) * Cch + kk + koff];
    a.y = w_red[(m0 + row) * Cch + kk + koff + 1];
#pragma unroll
    for (int j = 0; j < 4; ++j) {
      int col = n0 + j * 16 + row;
      v2f bb;
      bb.x = X[(kk + koff) * Pix + col];
      bb.y = X[(kk + koff + 1) * Pix + col];
      acc[j] = wmma4(a, bb, acc[j]);
    }
  }
  float* D = red + (size_t)b * CR * Pix;
#pragma unroll
  for (int j = 0; j < 4; ++j)
#pragma unroll
    for (int i = 0; i < 8; ++i) {
      int m = m0 + i + 8 * half;
      D[m * Pix + n0 + j * 16 + row] = acc[j][i] + b_red[m];
    }
}

// ---------------------------------------------------------------- K2: u,lam = W @ reduced
// Computed as D(P x C) = reduced^T x W^T so stores are pixel-major: u_t[b][p][c]
__global__ void k_expand(const float* __restrict__ red, const float* __restrict__ w_u,
                         const float* __restrict__ b_u, const float* __restrict__ w_lam,
                         const float* __restrict__ b_lam, float* __restrict__ u_t,
                         float* __restrict__ lam_t) {
  const int lane = threadIdx.x & 31;
  const int wid  = blockIdx.x * (blockDim.x >> 5) + (threadIdx.x >> 5); // 0..8191
  const int b    = wid >> 10;
  const int rem  = wid & 1023;
  const int mt   = rem >> 2;     // pixel tile 0..255
  const int nb   = rem & 3;      // 64-channel block
  const int row = lane & 15, half = lane >> 4, koff = half * 2;
  const int m0 = mt * 16;        // pixel base
  const int n0 = nb * 64;        // channel base
  const float* R = red + (size_t)b * CR * Pix;
  v8f accU[4] = {};
  v8f accL[4] = {};
  for (int kk = 0; kk < CR; kk += 4) {
    v2f a;                                   // A[M=pixel][K=cr]
    a.x = R[(kk + koff) * Pix + m0 + row];
    a.y = R[(kk + koff + 1) * Pix + m0 + row];
#pragma unroll
    for (int j = 0; j < 4; ++j) {
      int ch = n0 + j * 16 + row;
      v2f bu, bl;                            // B[K=cr][N=ch] = W[ch][cr]
      bu.x = w_u[ch * CR + kk + koff];
      bu.y = w_u[ch * CR + kk + koff + 1];
      bl.x = w_lam[ch * CR + kk + koff];
      bl.y = w_lam[ch * CR + kk + koff + 1];
      accU[j] = wmma4(a, bu, accU[j]);
      accL[j] = wmma4(a, bl, accL[j]);
    }
  }
#pragma unroll
  for (int j = 0; j < 4; ++j) {
    int ch = n0 + j * 16 + row;
    float biu = b_u[ch], bil = b_lam[ch];
#pragma unroll
    for (int i = 0; i < 8; ++i) {
      int p = m0 + i + 8 * half;
      size_t idx = ((size_t)b * Pix + p) * Cch + ch;
      u_t[idx]   = accU[j][i] + biu;
      lam_t[idx] = accL[j][i] + bil;
    }
  }
}

// ---------------------------------------------------------------- K2b: band weights -> R
__global__ void k_rcoef(const float* __restrict__ red, const float* __restrict__ w_wt,
                        const float* __restrict__ b_wt, float* __restrict__ Rg) {
  int idx = blockIdx.x * blockDim.x + threadIdx.x;   // 32768
  int b = idx >> 12, p = idx & 4095;
  const float* R = red + (size_t)b * CR * Pix + p;
  float s0 = b_wt[0], s1 = b_wt[1], s2 = b_wt[2];
  for (int k = 0; k < CR; ++k) {
    float rv = R[k * Pix];
    s0 = fmaf(w_wt[k],        rv, s0);
    s1 = fmaf(w_wt[64 + k],   rv, s1);
    s2 = fmaf(w_wt[128 + k],  rv, s2);
  }
  s0 = 1.f / (1.f + expf(-s0));
  s1 = 1.f / (1.f + expf(-s1));
  s2 = 1.f / (1.f + expf(-s2));
  float den = fmaxf(s0 + s1 + s2, 1e-6f);
  int w = p & 63;
  float num = s1 + (w >= 1 ? s0 : 0.f) + (w <= 62 ? s2 : 0.f);
  Rg[idx] = num / den;
}

// ---------------------------------------------------------------- K3: dual scan + folded merge
// One workgroup (256 thr = 8 waves) per (b,h). LDS: R row, staged x slice
// (async Global->LDS DMA when available), transposed O_A/O_B tiles (256ch x 64w)
// feeding the WMMA merge.
#define XPITCH 65
#define OPITCH 260
#define SMEM_FLOATS (64 + 256 * XPITCH + 2 * 64 * OPITCH)

__global__ void k_scan_merge(const float* __restrict__ x, const float* __restrict__ lam_t,
                             const float* __restrict__ u_t, const float* __restrict__ Rg,
                             const float* __restrict__ WA, const float* __restrict__ WB,
                             const float* __restrict__ b_merge, float* __restrict__ out) {
  extern __shared__ float smem[];
  float* Rs  = smem;                       // 64
  float* xs  = smem + 64;                  // 256 x XPITCH
  float* OAt = xs + 256 * XPITCH;          // 64 x OPITCH  (O_A[c][w] at [w][c])
  float* OBt = OAt + 64 * OPITCH;          // 64 x OPITCH

  const int b = blockIdx.x >> 6;
  const int h = blockIdx.x & 63;
  const int tid = threadIdx.x;

  if (tid < 64) Rs[tid] = Rg[(b << 12) + (h << 6) + tid];

  // Stage x[b][c][h*64 + 0..63] into LDS: one 16B transfer per (c, q) pair.
#if defined(USE_ASYNC_LDS)
#pragma unroll
  for (int t = tid; t < 256 * 16; t += 256) {
    int c = t >> 4, q = t & 15;
    v4i* g = (v4i*)(x + ((size_t)(b * Cch + c)) * Pix + h * 64 + q * 4);
    v4i* l = (v4i*)(xs + c * XPITCH + q * 4);
    __builtin_amdgcn_global_load_async_to_lds_b128(g, l, 0, 0);
  }
#if __has_builtin(__builtin_amdgcn_s_wait_asynccnt)
  __builtin_amdgcn_s_wait_asynccnt(0);
#else
  asm volatile("s_wait_asynccnt 0" ::: "memory");
#endif
#else
  for (int t = tid; t < 256 * 16; t += 256) {
    int c = t >> 4, q = t & 15;
    const float4 v = *(const float4*)(x + ((size_t)(b * Cch + c)) * Pix + h * 64 + q * 4);
    float* dst = xs + c * XPITCH + q * 4;
    dst[0] = v.x; dst[1] = v.y; dst[2] = v.z; dst[3] = v.w;
  }
#endif
  __syncthreads();

  // ---- Phase 1: H1/H2 scans, thread-per-channel, 64 serial steps along w.
  {
    const int c = tid;
    int lamA = (b * Pix + h) * Cch + c;                     // p = w*64+h    : +16384/step
    int lamB = (b * Pix + (63 * 64 + (63 - h))) * Cch + c;  // p=(63-w)*64+63-h : -16384/step
    int uA   = (b * Pix + h * 64) * Cch + c;                // p = h*64+w    : +256/step
    int uB   = (b * Pix + (63 - h) * 64) * Cch + c;         // p=(63-h)*64+w : +256/step
    float h1 = 0.f, h2 = 0.f;
    for (int w = 0; w < 64; ++w) {
      float r  = Rs[w];
      float xv = xs[c * XPITCH + w];
      // prefetch next step's lambda lines (stride 64KB walk)
      __builtin_prefetch(&lam_t[lamA + 64 * Cch], 0, 0);
      __builtin_prefetch(&lam_t[lamB - 64 * Cch], 0, 0);
      h1 = fmaf(h1, r, lam_t[lamA] * xv);
      h2 = fmaf(h2, r, lam_t[lamB] * xv);
      OAt[w * OPITCH + c] = h1 * u_t[uA];
      OBt[w * OPITCH + c] = h2 * u_t[uB];
      lamA += 64 * Cch; lamB -= 64 * Cch; uA += Cch; uB += Cch;
    }
  }
  __syncthreads();

  // ---- Phase 2: out[:,h,:] = WA @ O_A + WB @ O_B + b_merge  (256x64, K=256 twice)
  const int lane = tid & 31, wv = tid >> 5;
  const int row = lane & 15, half = lane >> 4, koff = 2 * half;
#pragma unroll
  for (int mi = 0; mi < 2; ++mi) {
    const int m0 = (wv * 2 + mi) * 16;       // output-channel tile base
    v8f acc[4] = {};
    for (int kk = 0; kk < 256; kk += 4) {
      v2f a;
      a.x = WA[(m0 + row) * 256 + kk + koff];
      a.y = WA[(m0 + row) * 256 + kk + koff + 1];
#pragma unroll
      for (int j = 0; j < 4; ++j) {
        int wcol = j * 16 + row;
        v2f bb;
        bb.x = OAt[wcol * OPITCH + kk + koff];
        bb.y = OAt[wcol * OPITCH + kk + koff + 1];
        acc[j] = wmma4(a, bb, acc[j]);
      }
    }
    for (int kk = 0; kk < 256; kk += 4) {
      v2f a;
      a.x = WB[(m0 + row) * 256 + kk + koff];
      a.y = WB[(m0 + row) * 256 + kk + koff + 1];
#pragma unroll
      for (int j = 0; j < 4; ++j) {
        int wcol = j * 16 + row;
        v2f bb;
        bb.x = OBt[wcol * OPITCH + kk + koff];
        bb.y = OBt[wcol * OPITCH + kk + koff + 1];
        acc[j] = wmma4(a, bb, acc[j]);
      }
    }
#pragma unroll
    for (int j = 0; j < 4; ++j)
#pragma unroll
      for (int i = 0; i < 8; ++i) {
        int o = m0 + i + 8 * half;
        out[((size_t)b * Cch + o) * Pix + h * 64 + j * 16 + row] =
            acc[j][i] + b_merge[o];
      }
  }
}

// ---------------------------------------------------------------- launch
extern "C" void kernel_launch(void* const* d_in, const int* in_sizes, int n_in,
                              void* d_out, int out_size, void* d_ws, size_t ws_size,
                              hipStream_t stream) {
  const float* x       = (const float*)d_in[0];
  const float* w_red   = (const float*)d_in[1];
  const float* b_red   = (const float*)d_in[2];
  const float* w_u     = (const float*)d_in[3];
  const float* b_u     = (const float*)d_in[4];
  const float* w_lam   = (const float*)d_in[5];
  const float* b_lam   = (const float*)d_in[6];
  const float* w_wt    = (const float*)d_in[7];
  const float* b_wt    = (const float*)d_in[8];
  const float* w_merge = (const float*)d_in[9];
  const float* b_merge = (const float*)d_in[10];
  float* out = (float*)d_out;

  float* ws    = (float*)d_ws;
  float* WA    = ws;                       // 65536
  float* WB    = WA + 65536;               // 65536
  float* red   = WB + 65536;               // 8*64*4096   = 2097152
  float* u_t   = red + 2097152;            // 8*4096*256  = 8388608
  float* lam_t = u_t + 8388608;            // 8388608
  float* Rg    = lam_t + 8388608;          // 32768
  (void)ws_size; (void)in_sizes; (void)n_in; (void)out_size;

  const size_t smem = SMEM_FLOATS * sizeof(float);   // ~195 KB of 320 KB/WGP
  (void)hipFuncSetAttribute((const void*)k_scan_merge,
                            hipFuncAttributeMaxDynamicSharedMemorySize, (int)smem);

  k_prep_merge<<<256, 256, 0, stream>>>(w_merge, WA, WB);
  k_reduce<<<512, 128, 0, stream>>>(x, w_red, b_red, red);
  k_expand<<<2048, 128, 0, stream>>>(red, w_u, b_u, w_lam, b_lam, u_t, lam_t);
  k_rcoef<<<128, 256, 0, stream>>>(red, w_wt, b_wt, Rg);
  k_scan_merge<<<512, 256, smem, stream>>>(x, lam_t, u_t, Rg, WA, WB, b_merge, out);
}